// RelationalGNN_26963804684494
// MI455X (gfx1250) — compile-verified
//
#include <hip/hip_runtime.h>

#define NN 100000
#define NE 3200000
#define NR 8

typedef __attribute__((ext_vector_type(2))) float v2f;
typedef __attribute__((ext_vector_type(8))) float v8f;

// ---------------- utility kernels ----------------

__global__ void zero_f32(float* __restrict__ p, size_t n) {
    size_t i = (size_t)blockIdx.x * blockDim.x + threadIdx.x;
    size_t stride = (size_t)gridDim.x * blockDim.x;
    for (; i < n; i += stride) p[i] = 0.0f;
}

__global__ void count_edges(const int* __restrict__ dst, const int* __restrict__ et,
                            float* __restrict__ cnt) {
    int e = blockIdx.x * blockDim.x + threadIdx.x;
    if (e >= NE) return;
    atomicAdd(&cnt[(size_t)et[e] * NN + dst[e]], 1.0f);
}

__global__ void recip_clamp(float* __restrict__ c, int n) {
    int i = blockIdx.x * blockDim.x + threadIdx.x;
    if (i < n) c[i] = 1.0f / fmaxf(c[i], 1.0f);
}

// one thread per (edge, 4-feature chunk); float4 gather (L2-resident) +
// 4 global_atomic_add_f32 into the per-(relation,dst) accumulator.
template <int F>
__global__ void scatter_edges(const float* __restrict__ feat, const int* __restrict__ src,
                              const int* __restrict__ dst, const int* __restrict__ et,
                              float* __restrict__ agg) {
    constexpr int Q = F / 4;
    long long tid = (long long)blockIdx.x * blockDim.x + threadIdx.x;
    int e = (int)(tid / Q);
    int q = (int)(tid % Q);
    if (e >= NE) return;
    int s = src[e], d = dst[e], r = et[e];
    const float4 v = ((const float4*)(feat + (size_t)s * F))[q];
    float* ap = agg + ((size_t)r * NN + d) * F + (size_t)q * 4;
    atomicAdd(ap + 0, v.x);
    atomicAdd(ap + 1, v.y);
    atomicAdd(ap + 2, v.z);
    atomicAdd(ap + 3, v.w);
}

// ---------------- WMMA fragment loaders (V_WMMA_F32_16X16X4_F32) ----------------
// A (16x4 f32): lane L -> row (L&15), K pair = k0 + 2*(L>>4) + {0,1}
// per-row scale applied here (row is constant per lane -> exact).
__device__ __forceinline__ v2f a_frag(const float* __restrict__ M, int row, int stride,
                                      int k0, int kh, float scale) {
    const float* p = M + (size_t)row * stride + k0 + kh;
    v2f a; a.x = p[0] * scale; a.y = p[1] * scale; return a;
}
// B (4x16 f32): lane L -> col (L&15), K pair = k0 + 2*(L>>4) + {0,1}
__device__ __forceinline__ v2f b_frag(const float* __restrict__ W, int nout,
                                      int k0, int kh, int col) {
    v2f b;
    b.x = W[(size_t)(k0 + kh) * nout + col];
    b.y = W[(size_t)(k0 + kh + 1) * nout + col];
    return b;
}

// ---------------- layer 1: x[N,16] -> h[N,32], fused bias+ReLU ----------------

__global__ __launch_bounds__(256) void rgcn_layer1(
    const float* __restrict__ x, const float* __restrict__ agg, const float* __restrict__ inv,
    const float* __restrict__ W1, const float* __restrict__ root1, const float* __restrict__ b1,
    float* __restrict__ h) {
    int wid  = (blockIdx.x * blockDim.x + threadIdx.x) >> 5;  // one wave32 per 16-node tile
    int lane = threadIdx.x & 31;
    if (wid >= NN / 16) return;
    int m0   = wid * 16;
    int arow = m0 + (lane & 15);
    int kh   = (lane >> 4) << 1;
    int col  = lane & 15;

    v8f c0, c1;
    float bb0 = b1[col], bb1 = b1[16 + col];
#pragma unroll
    for (int v = 0; v < 8; ++v) { c0[v] = bb0; c1[v] = bb1; }

    // root term: X(16x16) @ root1(16x32)
#pragma unroll
    for (int k = 0; k < 16; k += 4) {
        v2f a  = a_frag(x, arow, 16, k, kh, 1.0f);
        v2f bA = b_frag(root1, 32, k, kh, col);
        v2f bB = b_frag(root1, 32, k, kh, col + 16);
        c0 = __builtin_amdgcn_wmma_f32_16x16x4_f32(false, a, false, bA, (short)0, c0, false, false);
        c1 = __builtin_amdgcn_wmma_f32_16x16x4_f32(false, a, false, bB, (short)0, c1, false, false);
    }
    // relation terms: (agg_r * inv_r)(16x16) @ W1_r(16x32)
    for (int r = 0; r < NR; ++r) {
        const float* ag = agg + (size_t)r * NN * 16;
        const float* Wr = W1 + (size_t)r * 16 * 32;
        float sc = inv[(size_t)r * NN + arow];
#pragma unroll
        for (int k = 0; k < 16; k += 4) {
            v2f a  = a_frag(ag, arow, 16, k, kh, sc);
            v2f bA = b_frag(Wr, 32, k, kh, col);
            v2f bB = b_frag(Wr, 32, k, kh, col + 16);
            c0 = __builtin_amdgcn_wmma_f32_16x16x4_f32(false, a, false, bA, (short)0, c0, false, false);
            c1 = __builtin_amdgcn_wmma_f32_16x16x4_f32(false, a, false, bB, (short)0, c1, false, false);
        }
    }
    // C layout: VGPR v -> row m0 + v + 8*(lane>>4), col (lane&15); fuse ReLU.
    int rbase = m0 + ((lane >> 4) << 3);
#pragma unroll
    for (int v = 0; v < 8; ++v) {
        h[(size_t)(rbase + v) * 32 + col]      = fmaxf(c0[v], 0.0f);
        h[(size_t)(rbase + v) * 32 + 16 + col] = fmaxf(c1[v], 0.0f);
    }
}

// ---------------- layer 2: h[N,32] -> out[N,16], fused bias ----------------

__global__ __launch_bounds__(256) void rgcn_layer2(
    const float* __restrict__ h, const float* __restrict__ agg, const float* __restrict__ inv,
    const float* __restrict__ W2, const float* __restrict__ root2, const float* __restrict__ b2,
    float* __restrict__ out) {
    int wid  = (blockIdx.x * blockDim.x + threadIdx.x) >> 5;
    int lane = threadIdx.x & 31;
    if (wid >= NN / 16) return;
    int m0   = wid * 16;
    int arow = m0 + (lane & 15);
    int kh   = (lane >> 4) << 1;
    int col  = lane & 15;

    v8f c;
    float bb = b2[col];
#pragma unroll
    for (int v = 0; v < 8; ++v) c[v] = bb;

#pragma unroll
    for (int k = 0; k < 32; k += 4) {
        v2f a = a_frag(h, arow, 32, k, kh, 1.0f);
        v2f b = b_frag(root2, 16, k, kh, col);
        c = __builtin_amdgcn_wmma_f32_16x16x4_f32(false, a, false, b, (short)0, c, false, false);
    }
    for (int r = 0; r < NR; ++r) {
        const float* ag = agg + (size_t)r * NN * 32;
        const float* Wr = W2 + (size_t)r * 32 * 16;
        float sc = inv[(size_t)r * NN + arow];
#pragma unroll
        for (int k = 0; k < 32; k += 4) {
            v2f a = a_frag(ag, arow, 32, k, kh, sc);
            v2f b = b_frag(Wr, 16, k, kh, col);
            c = __builtin_amdgcn_wmma_f32_16x16x4_f32(false, a, false, b, (short)0, c, false, false);
        }
    }
    int rbase = m0 + ((lane >> 4) << 3);
#pragma unroll
    for (int v = 0; v < 8; ++v)
        out[(size_t)(rbase + v) * 16 + col] = c[v];
}

// ---------------- launcher ----------------

extern "C" void kernel_launch(void* const* d_in, const int* in_sizes, int n_in,
                              void* d_out, int out_size, void* d_ws, size_t ws_size,
                              hipStream_t stream) {
    const float* x     = (const float*)d_in[0];
    const int*   eidx  = (const int*)d_in[1];   // [2, E]
    const int*   etype = (const int*)d_in[2];
    const float* W1    = (const float*)d_in[3];
    const float* root1 = (const float*)d_in[4];
    const float* b1    = (const float*)d_in[5];
    const float* W2    = (const float*)d_in[6];
    const float* root2 = (const float*)d_in[7];
    const float* b2    = (const float*)d_in[8];
    float* out = (float*)d_out;

    const int* src = eidx;
    const int* dst = eidx + NE;

    // workspace: agg[8*N*32] | inv[8*N] | h[N*32]  (~118 MB, L2-resident on MI455X)
    float* agg = (float*)d_ws;
    float* inv = agg + (size_t)NR * NN * 32;
    float* h   = inv + (size_t)NR * NN;

    const int B = 256;

    // per-(relation,dst) edge counts -> 1/max(cnt,1); shared by both layers
    zero_f32<<<2048, B, 0, stream>>>(inv, (size_t)NR * NN);
    count_edges<<<(NE + B - 1) / B, B, 0, stream>>>(dst, etype, inv);
    recip_clamp<<<(NR * NN + B - 1) / B, B, 0, stream>>>(inv, NR * NN);

    // layer 1
    zero_f32<<<4096, B, 0, stream>>>(agg, (size_t)NR * NN * 16);
    {
        long long nth = (long long)NE * 4;
        scatter_edges<16><<<(int)((nth + B - 1) / B), B, 0, stream>>>(x, src, dst, etype, agg);
    }
    rgcn_layer1<<<((NN / 16) * 32 + B - 1) / B, B, 0, stream>>>(x, agg, inv, W1, root1, b1, h);

    // layer 2
    zero_f32<<<4096, B, 0, stream>>>(agg, (size_t)NR * NN * 32);
    {
        long long nth = (long long)NE * 8;
        scatter_edges<32><<<(int)((nth + B - 1) / B), B, 0, stream>>>(h, src, dst, etype, agg);
    }
    rgcn_layer2<<<((NN / 16) * 32 + B - 1) / B, B, 0, stream>>>(h, agg, inv, W2, root2, b2, out);

    (void)in_sizes; (void)n_in; (void)out_size; (void)ws_size;
}